// imagecomputation_49323404427296
// MI455X (gfx1250) — compile-verified
//
#include <hip/hip_runtime.h>
#include <math.h>

// ---------------------------------------------------------------------------
// Frame interpolation (Super-SloMo style): flow-composed bilinear warp of two
// RGB images + visibility-weighted blend. Shapes fixed by reference:
//   B=8, H=512, W=512.  Output: (8,512,512,3) f32.
//
// Memory-bound kernel (~1 FLOP/byte, ~150 MB traffic -> ~6.5us at 23.3 TB/s).
// CDNA5 path used: async global<->LDS B128 copies (ASYNCcnt) to stage the
// strided streaming tensors (interp 20B/px, flow 8B/px, out 12B/px) through
// LDS as aligned 16B transactions. Gathers (bilinear taps) ride the normal
// vector path and hit the 192MB L2 (both images = 50MB total fit entirely).
// ---------------------------------------------------------------------------

#define HAS_ASYNC_LDS 0
#if defined(__has_builtin)
#if __has_builtin(__builtin_amdgcn_global_load_async_to_lds_b128) && \
    __has_builtin(__builtin_amdgcn_global_store_async_from_lds_b128) && \
    __has_builtin(__builtin_amdgcn_s_wait_asynccnt)
#undef HAS_ASYNC_LDS
#define HAS_ASYNC_LDS 1
#endif
#endif

namespace {

constexpr int TPB = 256;      // 8 wave32 waves per block
constexpr int IMG_W = 512;
constexpr int IMG_H = 512;
constexpr int IMG_HW = IMG_W * IMG_H;   // 2^18

#if HAS_ASYNC_LDS
// Builtin signature (from hipcc diagnostic): (int4 AS1*, int4 AS3*, imm, imm)
typedef int v4i __attribute__((vector_size(16)));
typedef __attribute__((address_space(1))) v4i* g_v4i_ptr;
typedef __attribute__((address_space(3))) v4i* l_v4i_ptr;
#endif

__device__ __forceinline__ int clampi(int v, int lo, int hi) {
  return v < lo ? lo : (v > hi ? hi : v);
}

// Bilinear sample of a (H,W,3) plane at (xs, ys) with edge clamping.
// One thread handles all 3 channels so weights/addresses are computed once.
__device__ __forceinline__ void bilinear3(const float* __restrict__ img,
                                          long long planeBase,
                                          float xs, float ys, float g[3]) {
  float x0f = floorf(xs);
  float y0f = floorf(ys);
  float wx = xs - x0f;
  float wy = ys - y0f;
  int x0 = clampi((int)x0f,     0, IMG_W - 1);
  int x1 = clampi((int)x0f + 1, 0, IMG_W - 1);
  int y0 = clampi((int)y0f,     0, IMG_H - 1);
  int y1 = clampi((int)y0f + 1, 0, IMG_H - 1);
  const float* p00 = img + planeBase + ((long long)y0 * IMG_W + x0) * 3;  // Ia
  const float* p01 = img + planeBase + ((long long)y0 * IMG_W + x1) * 3;  // Ic
  const float* p10 = img + planeBase + ((long long)y1 * IMG_W + x0) * 3;  // Ib
  const float* p11 = img + planeBase + ((long long)y1 * IMG_W + x1) * 3;  // Id
  float w00 = (1.f - wx) * (1.f - wy);
  float w01 = wx * (1.f - wy);
  float w10 = (1.f - wx) * wy;
  float w11 = wx * wy;
#pragma unroll
  for (int c = 0; c < 3; ++c) {
    g[c] = fmaf(w00, p00[c],
           fmaf(w01, p01[c],
           fmaf(w10, p10[c], w11 * p11[c])));
  }
}

__global__ __launch_bounds__(TPB)
void frame_interp_kernel(const float* __restrict__ tptr,
                         const float* __restrict__ I0,
                         const float* __restrict__ I1,
                         const float* __restrict__ interp,
                         const float* __restrict__ F0,
                         const float* __restrict__ F1,
                         float* __restrict__ out,
                         int npix) {
  const int tid  = (int)threadIdx.x;
  const int base = (int)blockIdx.x * TPB;
  const int idx  = base + tid;

#if HAS_ASYNC_LDS
  __shared__ float lds_itp[TPB * 5];  // 5120 B = 320 x 16B chunks
  __shared__ float lds_f0 [TPB * 2];  // 2048 B = 128 chunks
  __shared__ float lds_f1 [TPB * 2];  // 2048 B = 128 chunks
  __shared__ float lds_out[TPB * 3];  // 3072 B = 192 chunks
  {
    const int rem_px = npix - base;                     // full blocks here
    const char* gitp = (const char*)(interp + (long long)base * 5);
    const char* gf0  = (const char*)(F0 + (long long)base * 2);
    const char* gf1  = (const char*)(F1 + (long long)base * 2);
    char* litp = (char*)lds_itp;
    char* lf0  = (char*)lds_f0;
    char* lf1  = (char*)lds_f1;

    // interp tile: 320 chunks -> all 256 threads + first 64 take a second one
    if (tid * 16 < rem_px * 20) {
      __builtin_amdgcn_global_load_async_to_lds_b128(
          (g_v4i_ptr)(gitp + tid * 16),
          (l_v4i_ptr)(litp + tid * 16), 0, 0);
    }
    if (tid < 64 && (256 + tid) * 16 < rem_px * 20) {
      __builtin_amdgcn_global_load_async_to_lds_b128(
          (g_v4i_ptr)(gitp + (256 + tid) * 16),
          (l_v4i_ptr)(litp + (256 + tid) * 16), 0, 0);
    }
    // flow tiles: 128 chunks each
    if (tid < 128 && tid * 16 < rem_px * 8) {
      __builtin_amdgcn_global_load_async_to_lds_b128(
          (g_v4i_ptr)(gf0 + tid * 16),
          (l_v4i_ptr)(lf0 + tid * 16), 0, 0);
      __builtin_amdgcn_global_load_async_to_lds_b128(
          (g_v4i_ptr)(gf1 + tid * 16),
          (l_v4i_ptr)(lf1 + tid * 16), 0, 0);
    }
    __builtin_amdgcn_s_wait_asynccnt(0);  // this wave's async copies landed
    __syncthreads();                       // all waves' copies landed
  }
#endif

  float r0 = 0.f, r1 = 0.f, r2 = 0.f;
  if (idx < npix) {
#if HAS_ASYNC_LDS
    const float iv0 = lds_itp[tid * 5 + 0];
    const float iv1 = lds_itp[tid * 5 + 1];
    const float iv2 = lds_itp[tid * 5 + 2];
    const float iv3 = lds_itp[tid * 5 + 3];
    const float iv4 = lds_itp[tid * 5 + 4];
    const float f0x = lds_f0[tid * 2 + 0];
    const float f0y = lds_f0[tid * 2 + 1];
    const float f1x = lds_f1[tid * 2 + 0];
    const float f1y = lds_f1[tid * 2 + 1];
#else
    const float* ip = interp + (long long)idx * 5;
    const float iv0 = ip[0], iv1 = ip[1], iv2 = ip[2], iv3 = ip[3], iv4 = ip[4];
    const float f0x = F0[(long long)idx * 2 + 0];
    const float f0y = F0[(long long)idx * 2 + 1];
    const float f1x = F1[(long long)idx * 2 + 0];
    const float f1y = F1[(long long)idx * 2 + 1];
#endif
    const int b   = idx >> 18;               // / (512*512)
    const int rem = idx & (IMG_HW - 1);
    const float yc = (float)(rem >> 9);      // / 512
    const float xc = (float)(rem & (IMG_W - 1));

    const float tv  = tptr[b];
    const float Vt0 = 1.f / (1.f + expf(-iv4));

    float g0[3], g1[3];
    const long long plane = (long long)b * IMG_HW * 3;
    bilinear3(I0, plane, xc + (iv0 + f0x), yc + (iv1 + f0y), g0);
    bilinear3(I1, plane, xc + (iv2 + f1x), yc + (iv3 + f1y), g1);

    const float w0  = (1.f - tv) * Vt0;
    const float w1  = tv * (1.f - Vt0);
    const float inv = 1.f / (w0 + w1 + 1e-12f);
    r0 = (w0 * g0[0] + w1 * g1[0]) * inv;
    r1 = (w0 * g0[1] + w1 * g1[1]) * inv;
    r2 = (w0 * g0[2] + w1 * g1[2]) * inv;
  }

#if HAS_ASYNC_LDS
  lds_out[tid * 3 + 0] = r0;
  lds_out[tid * 3 + 1] = r1;
  lds_out[tid * 3 + 2] = r2;
  __syncthreads();
  // 192 x 16B async chunks: pure coalesced B128 store stream.
  if (tid < 192 && (long long)base * 12 + tid * 16 + 16 <= (long long)npix * 12) {
    char* gout = (char*)(out + (long long)base * 3);
    __builtin_amdgcn_global_store_async_from_lds_b128(
        (g_v4i_ptr)(gout + tid * 16),
        (l_v4i_ptr)((char*)lds_out + tid * 16), 0, 0);
  }
  // s_endpgm performs an implicit wait-idle; no explicit asynccnt wait needed.
#else
  if (idx < npix) {
    out[(long long)idx * 3 + 0] = r0;
    out[(long long)idx * 3 + 1] = r1;
    out[(long long)idx * 3 + 2] = r2;
  }
#endif
}

}  // namespace

extern "C" void kernel_launch(void* const* d_in, const int* in_sizes, int n_in,
                              void* d_out, int out_size, void* d_ws, size_t ws_size,
                              hipStream_t stream) {
  const float* t   = (const float*)d_in[0];  // (B,1,1,1)
  const float* I0  = (const float*)d_in[1];  // (B,H,W,3)
  const float* I1  = (const float*)d_in[2];  // (B,H,W,3)
  const float* itp = (const float*)d_in[3];  // (B,H,W,5)
  const float* F0  = (const float*)d_in[4];  // (B,H,W,2)
  const float* F1  = (const float*)d_in[5];  // (B,H,W,2)
  float* out = (float*)d_out;                // (B,H,W,3)

  const int npix   = out_size / 3;           // 2,097,152 for the reference shapes
  const int blocks = (npix + TPB - 1) / TPB; // 8192 blocks x 8 waves

  hipLaunchKernelGGL(frame_interp_kernel, dim3(blocks), dim3(TPB), 0, stream,
                     t, I0, I1, itp, F0, F1, out, npix);
}